// GaussianKernelLayer_25503515803935
// MI455X (gfx1250) — compile-verified
//
#include <hip/hip_runtime.h>
#include <math.h>

typedef float v2f __attribute__((ext_vector_type(2)));
typedef float v8f __attribute__((ext_vector_type(8)));

#define WAVES_PER_BLOCK 8
#define BLOCK_ROWS (WAVES_PER_BLOCK * 16)

// ---------------------------------------------------------------------------
// Prep kernel (1 workgroup): fp64 mean of coefficients, fold norm_const into
// centered weights, down-convert centers to fp32, precompute |c|^2 (fp64 acc).
// ---------------------------------------------------------------------------
__global__ __launch_bounds__(256) void gk_prep(const double* __restrict__ centers,
                                               const double* __restrict__ coef,
                                               float* __restrict__ cf,   // [M,32] fp32 centers
                                               float* __restrict__ csq,  // [M]   |c|^2
                                               float* __restrict__ wcf,  // [M]   norm_const*(w - mean)
                                               int M)
{
    __shared__ double red[256];
    const int t = threadIdx.x;

    double s = 0.0;
    for (int m = t; m < M; m += 256) s += coef[m];
    red[t] = s;
    __syncthreads();
    for (int off = 128; off > 0; off >>= 1) {
        if (t < off) red[t] += red[t + off];
        __syncthreads();
    }
    const double mean = red[0] / (double)M;
    // 1 / ((2*pi)^(D/2) * sigma^D), D=32, sigma=4
    const double nc = 1.0 / (pow(2.0 * 3.14159265358979323846, 16.0) * pow(4.0, 32.0));

    for (int m = t; m < M; m += 256) {
        wcf[m] = (float)((coef[m] - mean) * nc);
        double ss = 0.0;
        #pragma unroll
        for (int d = 0; d < 32; ++d) {
            const double c = centers[(size_t)m * 32 + d];
            cf[(size_t)m * 32 + d] = (float)c;
            ss += c * c;
        }
        csq[m] = (float)ss;
    }
}

// ---------------------------------------------------------------------------
// Main fused kernel: per wave, a 16-row x tile in WMMA-A f32 layout; sweep all
// centers in 16-column tiles with 8x V_WMMA_F32_16X16X4_F32 (K=32), fused
// Gaussian epilogue, cross-lane reduction. Never materializes the [N,M] matrix.
// ---------------------------------------------------------------------------
__global__ __launch_bounds__(256) void gk_main(const float* __restrict__ x,
                                               const float* __restrict__ cf,
                                               const float* __restrict__ csq,
                                               const float* __restrict__ wcf,
                                               float* __restrict__ out,
                                               int M)
{
    const int wave = threadIdx.x >> 5;
    const int lane = threadIdx.x & 31;
    const int hi   = lane >> 4;       // which half of the wave
    const int l16  = lane & 15;
    const int n0   = blockIdx.x * BLOCK_ROWS + wave * 16;
    const int row  = n0 + l16;

    const float* __restrict__ xr = x + (size_t)row * 32;

    // |x_row|^2 : lane L (and L+16) computes row l16's sum of squares
    float xsq = 0.0f;
    #pragma unroll
    for (int d = 0; d < 32; ++d) xsq += xr[d] * xr[d];

    // Per lane we need |x|^2 for the 8 C/D rows it covers: r + 8*hi
    float xsr[8];
    #pragma unroll
    for (int r = 0; r < 8; ++r) xsr[r] = __shfl(xsq, r + 8 * hi, 32);

    // A-matrix 16x4 f32 layout (ISA 7.12.2): lanes 0-15 hold K=4s..4s+1,
    // lanes 16-31 hold K=4s+2..4s+3, M = lane&15. 8 slices cover K=32.
    v2f a[8];
    #pragma unroll
    for (int s = 0; s < 8; ++s) {
        const int k = 4 * s + 2 * hi;
        a[s] = *(const v2f*)(xr + k);
    }

    float sum[8];
    #pragma unroll
    for (int r = 0; r < 8; ++r) sum[r] = 0.0f;

    const float kscale = -0.03125f;   // -0.5 / sigma^2, sigma = 4

    for (int m0 = 0; m0 < M; m0 += 16) {
        const int m = m0 + l16;
        const float* __restrict__ cr = cf + (size_t)m * 32;

        // B-matrix 4x16 (mirror of A): N = lane&15, same K striping
        v2f b[8];
        #pragma unroll
        for (int s = 0; s < 8; ++s) {
            const int k = 4 * s + 2 * hi;
            b[s] = *(const v2f*)(cr + k);
        }

        if (m0 + 16 < M)
            __builtin_prefetch(cf + (size_t)(m + 16) * 32, 0, 0);  // global_prefetch_b8

        // D = A x B + C, chained over 8 K-slices => full fp32 x.c^T, K=32
        v8f acc = {0.f, 0.f, 0.f, 0.f, 0.f, 0.f, 0.f, 0.f};
        #pragma unroll
        for (int s = 0; s < 8; ++s) {
            acc = __builtin_amdgcn_wmma_f32_16x16x4_f32(
                      false, a[s], false, b[s], (short)0, acc, false, false);
        }

        // Fused Gaussian epilogue. D-layout: VGPR r, lane L -> row r+8*hi,
        // col lane&15. csq/wcf indexed by this lane's center column.
        const float cs = csq[m];
        const float w  = wcf[m];
        #pragma unroll
        for (int r = 0; r < 8; ++r) {
            const float dist = xsr[r] + cs - 2.0f * acc[r];
            sum[r] += w * __expf(kscale * dist);
        }
    }

    // Reduce over the 16 columns held by each half-wave
    #pragma unroll
    for (int r = 0; r < 8; ++r) {
        float v = sum[r];
        v += __shfl_xor(v, 1, 32);
        v += __shfl_xor(v, 2, 32);
        v += __shfl_xor(v, 4, 32);
        v += __shfl_xor(v, 8, 32);
        sum[r] = v;
    }
    if (l16 == 0) {
        #pragma unroll
        for (int r = 0; r < 8; ++r)
            out[n0 + 8 * hi + r] = sum[r];
    }
}

// ---------------------------------------------------------------------------
extern "C" void kernel_launch(void* const* d_in, const int* in_sizes, int n_in,
                              void* d_out, int out_size, void* d_ws, size_t ws_size,
                              hipStream_t stream) {
    const float*  x       = (const float*)d_in[0];   // [N,32] fp32
    const double* centers = (const double*)d_in[1];  // [M,32] fp64
    const double* coef    = (const double*)d_in[2];  // [M]    fp64
    float* out = (float*)d_out;                      // [N]

    const int N = in_sizes[0] / 32;
    const int M = in_sizes[2];

    // Workspace layout: fp32 centers [M*32], |c|^2 [M], centered weights [M]
    float* cf  = (float*)d_ws;
    float* csq = cf + (size_t)M * 32;
    float* wcf = csq + M;

    gk_prep<<<1, 256, 0, stream>>>(centers, coef, cf, csq, wcf, M);
    gk_main<<<N / BLOCK_ROWS, 256, 0, stream>>>(x, cf, csq, wcf, out, M);
}